// ContrastiveLoss_18279380811979
// MI455X (gfx1250) — compile-verified
//
#include <hip/hip_runtime.h>
#include <hip/hip_bf16.h>

typedef __attribute__((ext_vector_type(2))) float v2f;
typedef __attribute__((ext_vector_type(8))) float v8f;

#define B_   4
#define M_   50
#define C_   256
#define H_   100
#define W_   352
#define N_   (B_ * M_)     // 200
#define HW_  (H_ * W_)     // 35200
#define NPAD 208           // 13 * 16
#define TAU_ 0.07f

// ---------------------------------------------------------------------------
// Kernel 1: per-object bounding box + popcount of each (b, m) mask.
// rects[obj*5 + {0..4}] = ymin, ymax, xmin, xmax, count
// ---------------------------------------------------------------------------
__global__ void bbox_kernel(const unsigned char* __restrict__ mask,
                            int* __restrict__ rects) {
    __shared__ int s0[256], s1[256], s2[256], s3[256], s4[256];
    const int obj = blockIdx.x;
    const unsigned char* m = mask + (size_t)obj * HW_;
    int ymin = H_, ymax = -1, xmin = W_, xmax = -1, cnt = 0;
    for (int p = threadIdx.x; p < HW_; p += blockDim.x) {
        if (m[p]) {
            int y = p / W_;
            int x = p - y * W_;
            ymin = min(ymin, y); ymax = max(ymax, y);
            xmin = min(xmin, x); xmax = max(xmax, x);
            ++cnt;
        }
    }
    const int t = threadIdx.x;
    s0[t] = ymin; s1[t] = ymax; s2[t] = xmin; s3[t] = xmax; s4[t] = cnt;
    __syncthreads();
    for (int st = 128; st > 0; st >>= 1) {
        if (t < st) {
            s0[t] = min(s0[t], s0[t + st]);
            s1[t] = max(s1[t], s1[t + st]);
            s2[t] = min(s2[t], s2[t + st]);
            s3[t] = max(s3[t], s3[t + st]);
            s4[t] += s4[t + st];
        }
        __syncthreads();
    }
    if (t == 0) {
        int* r = rects + obj * 5;
        r[0] = s0[0]; r[1] = s1[0]; r[2] = s2[0]; r[3] = s3[0]; r[4] = s4[0];
    }
}

// ---------------------------------------------------------------------------
// Kernel 2: segment means over the rectangle intersection, then L2 normalize.
// Object i pairs rect A = rects[i]  (mask[i//M, i%M], also the count source)
// with rect B = rects[(i%b)*M + i//b] (mask[i%b, i//b]); features of batch i%b.
// One block per object, one thread per channel (C_ == 256 == blockDim.x).
// Grid is NPAD blocks: blocks i >= N_ zero-fill the WMMA padding rows so the
// GEMM kernel can load unconditionally (no EXEC-masked scalar loads).
// ---------------------------------------------------------------------------
__global__ void segmean_kernel(const float* __restrict__ fq,
                               const float* __restrict__ fk,
                               const int* __restrict__ rects,
                               float* __restrict__ nq,
                               float* __restrict__ nk,
                               float* __restrict__ pad) {
    __shared__ float sh[256];
    const int i  = blockIdx.x;
    const int ch = threadIdx.x;

    if (i >= N_) {                 // uniform per-block early path: zero padding
        nq[(size_t)i * C_ + ch] = 0.f;
        nk[(size_t)i * C_ + ch] = 0.f;
        return;
    }

    const int* rA = rects + i * 5;
    const int  bB = i % B_;
    const int  mB = i / B_;
    const int* rB = rects + (bB * M_ + mB) * 5;

    const int y0 = max(rA[0], rB[0]);
    const int y1 = min(rA[1], rB[1]);
    const int x0 = max(rA[2], rB[2]);
    const int x1 = min(rA[3], rB[3]);
    const float cnt = (float)max(rA[4], 1);   // reference guarantees cnt > 0

    const int r = i % B_;                     // feature batch
    const float* q = fq + ((size_t)r * C_ + ch) * HW_;
    const float* k = fk + ((size_t)r * C_ + ch) * HW_;

    float sq = 0.f, sk = 0.f;
    for (int y = y0; y <= y1; ++y) {
        const int base = y * W_;
        for (int x = x0; x <= x1; ++x) {
            sq += q[base + x];
            sk += k[base + x];
        }
    }
    sq /= cnt;
    sk /= cnt;

    if (ch == 0) pad[i] = (sk != 0.f) ? 1.f : 0.f;  // validity from raw mean

    // L2 norm of sq over channels
    sh[ch] = sq * sq; __syncthreads();
    for (int st = 128; st > 0; st >>= 1) {
        if (ch < st) sh[ch] += sh[ch + st];
        __syncthreads();
    }
    const float nrmq = fmaxf(sqrtf(sh[0]), 1e-12f);
    __syncthreads();

    // L2 norm of sk over channels
    sh[ch] = sk * sk; __syncthreads();
    for (int st = 128; st > 0; st >>= 1) {
        if (ch < st) sh[ch] += sh[ch + st];
        __syncthreads();
    }
    const float nrmk = fmaxf(sqrtf(sh[0]), 1e-12f);

    nq[(size_t)i * C_ + ch] = sq / nrmq;
    nk[(size_t)i * C_ + ch] = sk / nrmk;
}

// ---------------------------------------------------------------------------
// Kernel 3: logits = (nk @ nq.T) / tau via V_WMMA_F32_16X16X4_F32.
// One wave (32 threads) per 16x16 tile; K = 256 in 64 steps of 4.
// A layout (16x4 f32): lane = m + 16*kh, VGPR0/1 = K (2*kh), (2*kh+1).
// B layout mirrors A with N in place of M, so the same addressing reads nq.
// C/D layout: VGPR v holds row M = v + 8*kh, column N = lane & 15.
// Padding rows of nq/nk are pre-zeroed, so loads are unconditional b64.
// ---------------------------------------------------------------------------
__global__ void logits_wmma_kernel(const float* __restrict__ nk,
                                   const float* __restrict__ nq,
                                   float* __restrict__ logits) {
    const int i0   = blockIdx.y * 16;          // row tile (nk rows)
    const int j0   = blockIdx.x * 16;          // col tile (nq rows)
    const int lane = threadIdx.x;              // 0..31 (wave32)
    const int mn   = lane & 15;
    const int kh   = lane >> 4;                // K half-select

    const v2f* pa = (const v2f*)(nk + (size_t)(i0 + mn) * C_);
    const v2f* pb = (const v2f*)(nq + (size_t)(j0 + mn) * C_);

    v8f c = {};
#pragma unroll 4
    for (int kb = 0; kb < C_; kb += 4) {
        const int e = (kb >> 1) + kh;          // v2f element index
        v2f a = pa[e];                         // global_load_b64, coalesced
        v2f b = pb[e];
        c = __builtin_amdgcn_wmma_f32_16x16x4_f32(
                /*neg_a=*/false, a, /*neg_b=*/false, b,
                /*c_mod=*/(short)0, c, /*reuse_a=*/false, /*reuse_b=*/false);
    }

    const float inv_tau = 1.0f / TAU_;
#pragma unroll
    for (int v = 0; v < 8; ++v) {
        const int row = i0 + v + 8 * kh;
        const int col = j0 + mn;
        logits[(size_t)row * NPAD + col] = c[v] * inv_tau;
    }
}

// ---------------------------------------------------------------------------
// Kernel 4: per-row logsumexp, cross-entropy with diagonal labels,
// pad-weighted mean. One block; one thread per row.
// ---------------------------------------------------------------------------
__global__ void loss_kernel(const float* __restrict__ logits,
                            const float* __restrict__ pad,
                            float* __restrict__ out) {
    __shared__ float sCe[256], sPad[256];
    const int t = threadIdx.x;
    float ce = 0.f, pf = 0.f;
    if (t < N_) {
        const float* row = logits + (size_t)t * NPAD;
        float mx = -1e30f;
        for (int j = 0; j < N_; ++j) mx = fmaxf(mx, row[j]);
        float s = 0.f;
        for (int j = 0; j < N_; ++j) s += expf(row[j] - mx);
        const float lse = mx + logf(s);
        pf = pad[t];
        ce = (lse - row[t]) * pf;
    }
    sCe[t] = ce; sPad[t] = pf;
    __syncthreads();
    for (int st = 128; st > 0; st >>= 1) {
        if (t < st) { sCe[t] += sCe[t + st]; sPad[t] += sPad[t + st]; }
        __syncthreads();
    }
    if (t == 0) out[0] = sCe[0] / fmaxf(sPad[0], 1.f);
}

// ---------------------------------------------------------------------------
extern "C" void kernel_launch(void* const* d_in, const int* in_sizes, int n_in,
                              void* d_out, int out_size, void* d_ws, size_t ws_size,
                              hipStream_t stream) {
    const float* fq          = (const float*)d_in[0];
    const float* fk          = (const float*)d_in[1];
    const unsigned char* msk = (const unsigned char*)d_in[2];  // bool tensor
    float* out               = (float*)d_out;

    // Workspace layout (all offsets 256B-aligned; ~590 KB total)
    char* ws = (char*)d_ws;
    size_t off = 0;
    int* rects = (int*)(ws + off);      off += ((size_t)N_ * 5 * sizeof(int) + 255) & ~(size_t)255;
    float* nq  = (float*)(ws + off);    off += (size_t)NPAD * C_ * sizeof(float);
    float* nk  = (float*)(ws + off);    off += (size_t)NPAD * C_ * sizeof(float);
    float* pad = (float*)(ws + off);    off += ((size_t)N_ * sizeof(float) + 255) & ~(size_t)255;
    float* lgt = (float*)(ws + off);    off += (size_t)NPAD * NPAD * sizeof(float);

    bbox_kernel<<<N_, 256, 0, stream>>>(msk, rects);
    segmean_kernel<<<NPAD, 256, 0, stream>>>(fq, fk, rects, nq, nk, pad);
    dim3 g3(NPAD / 16, NPAD / 16);               // 13 x 13 tiles, 1 wave each
    logits_wmma_kernel<<<g3, 32, 0, stream>>>(nk, nq, lgt);
    loss_kernel<<<1, 256, 0, stream>>>(lgt, pad, out);
}